// PatchSampleF_7464653160556
// MI455X (gfx1250) — compile-verified
//
#include <hip/hip_runtime.h>
#include <math.h>
#include <float.h>

typedef __attribute__((ext_vector_type(2))) float v2f;
typedef __attribute__((ext_vector_type(8))) float v8f;

#define B_  2
#define C_  256
#define N3_ 4096
#define P_  64

// ---------------- Kernel 1: gather + L2 normalize (feat0..2 branches) -------
__global__ void gather_norm_kernel(const float* __restrict__ f,
                                   const int* __restrict__ pid,
                                   float* __restrict__ out,
                                   int Nsp) {
    __shared__ float red[C_];
    int b = blockIdx.x / P_;
    int p = blockIdx.x % P_;
    int c = threadIdx.x;
    int n = pid[p];
    float v = f[((size_t)(b * C_ + c)) * Nsp + n];
    red[c] = v * v;
    __syncthreads();
    for (int s = C_ / 2; s > 0; s >>= 1) {
        if (c < s) red[c] += red[c + s];
        __syncthreads();
    }
    float norm = sqrtf(red[0]) + 1e-7f;
    out[(size_t)(b * P_ + p) * C_ + c] = v / norm;
}

// ---------------- Kernel 2: fp32 WMMA gram rows + online softmax entropy ----
// X = feat3[b] viewed as [C, N] row-major.  dots[r][c] = sum_k X[k][r]*X[k][c].
// Each wave processes 16x64 super-tiles (4 WMMA tiles) so one A fragment feeds
// 4 WMMAs and the shuffle reduction runs once per 64 columns.
__global__ void entropy_kernel(const float* __restrict__ feat3,
                               float* __restrict__ entg,
                               float* __restrict__ lseg) {
    __shared__ float As[C_ * 16];            // As[c*16 + m] = X[c][rowbase+m]
    __shared__ float smemStats[4][16][3];
    const int b = blockIdx.x / (N3_ / 16);
    const int rt = blockIdx.x % (N3_ / 16);
    const int rowbase = rt * 16;
    const float* X = feat3 + (size_t)b * C_ * N3_;

    for (int idx = threadIdx.x; idx < C_ * 16; idx += blockDim.x) {
        int c = idx >> 4, m = idx & 15;
        As[idx] = X[(size_t)c * N3_ + rowbase + m];
    }
    __syncthreads();

    const int wave = threadIdx.x >> 5;
    const int lane = threadIdx.x & 31;
    const int g  = lane >> 4;                // half-group (K offset 0 or 2)
    const int lh = lane & 15;

    float runM[8], runS[8], runW[8];
    for (int r = 0; r < 8; ++r) { runM[r] = -INFINITY; runS[r] = 0.f; runW[r] = 0.f; }

    for (int st = wave; st < N3_ / 64; st += 4) {
        const int colbase = st * 64;
        // speculative prefetch of the next super-tile's B panel
        {
            int nst = st + 4;
            if (nst < N3_ / 64) {
                const float* pf = X + nst * 64 + lane * 2;   // 32 lanes * 8B = 256B row seg
                #pragma unroll
                for (int kk = 0; kk < C_; kk += 16)
                    __builtin_prefetch(pf + (size_t)kk * N3_, 0, 1);
            }
        }
        const float* Xc = X + colbase + lh;
        v8f acc0 = {}, acc1 = {}, acc2 = {}, acc3 = {};
        for (int k = 0; k < C_; k += 4) {
            v2f a;
            a.x = As[(k + 2 * g) * 16 + lh];
            a.y = As[(k + 2 * g + 1) * 16 + lh];
            const float* r0 = Xc + (size_t)(k + 2 * g) * N3_;
            const float* r1 = Xc + (size_t)(k + 2 * g + 1) * N3_;
            v2f b0, b1, b2, b3;
            b0.x = r0[0];  b0.y = r1[0];
            b1.x = r0[16]; b1.y = r1[16];
            b2.x = r0[32]; b2.y = r1[32];
            b3.x = r0[48]; b3.y = r1[48];
            acc0 = __builtin_amdgcn_wmma_f32_16x16x4_f32(false, a, false, b0, (short)0, acc0, false, false);
            acc1 = __builtin_amdgcn_wmma_f32_16x16x4_f32(false, a, false, b1, (short)0, acc1, false, false);
            acc2 = __builtin_amdgcn_wmma_f32_16x16x4_f32(false, a, false, b2, (short)0, acc2, false, false);
            acc3 = __builtin_amdgcn_wmma_f32_16x16x4_f32(false, a, false, b3, (short)0, acc3, false, false);
        }
        // per-row online softmax stats across 64 columns at once
        for (int r = 0; r < 8; ++r) {
            float d0 = acc0[r], d1 = acc1[r], d2 = acc2[r], d3 = acc3[r];
            float mx = fmaxf(fmaxf(d0, d1), fmaxf(d2, d3));
            for (int m = 1; m < 16; m <<= 1) mx = fmaxf(mx, __shfl_xor(mx, m, 32));
            float e0 = __expf(d0 - mx), e1 = __expf(d1 - mx);
            float e2 = __expf(d2 - mx), e3 = __expf(d3 - mx);
            float se = (e0 + e1) + (e2 + e3);
            float sw = (e0 * d0 + e1 * d1) + (e2 * d2 + e3 * d3);
            for (int m = 1; m < 16; m <<= 1) {
                se += __shfl_xor(se, m, 32);
                sw += __shfl_xor(sw, m, 32);
            }
            if (mx > runM[r]) {
                float sc = __expf(runM[r] - mx);
                runS[r] = runS[r] * sc + se;
                runW[r] = runW[r] * sc + sw;
                runM[r] = mx;
            } else {
                float sc = __expf(mx - runM[r]);
                runS[r] += se * sc;
                runW[r] += sw * sc;
            }
        }
    }
    if (lh == 0) {                            // lane 0 -> rows 0..7, lane 16 -> 8..15
        for (int r = 0; r < 8; ++r) {
            int row16 = r + 8 * g;
            smemStats[wave][row16][0] = runM[r];
            smemStats[wave][row16][1] = runS[r];
            smemStats[wave][row16][2] = runW[r];
        }
    }
    __syncthreads();
    if (threadIdx.x < 16) {
        int t = threadIdx.x;
        float M = -INFINITY, S = 0.f, W = 0.f;
        for (int w = 0; w < 4; ++w) {
            float m = smemStats[w][t][0], s = smemStats[w][t][1], ww = smemStats[w][t][2];
            if (m > M) {
                float sc = __expf(M - m);
                S = S * sc + s; W = W * sc + ww; M = m;
            } else {
                float sc = __expf(m - M);
                S += s * sc; W += ww * sc;
            }
        }
        float lse = M + logf(S);
        entg[(size_t)b * N3_ + rowbase + t] = lse - W / S;   // entropy
        lseg[(size_t)b * N3_ + rowbase + t] = lse;
    }
}

// ---------------- Kernel 3: 64 smallest entropies, ascending, per batch -----
__global__ void select_kernel(const float* __restrict__ entg,
                              int* __restrict__ sel) {
    __shared__ float se[N3_];
    __shared__ float rv[256];
    __shared__ int   ri[256];
    int b = blockIdx.x;
    int tid = threadIdx.x;
    for (int i = tid; i < N3_; i += 256) se[i] = entg[(size_t)b * N3_ + i];
    __syncthreads();
    for (int j = 0; j < P_; ++j) {
        float best = FLT_MAX; int bi = N3_;
        for (int i = tid; i < N3_; i += 256) {
            float v = se[i];
            if (v < best || (v == best && i < bi)) { best = v; bi = i; }
        }
        rv[tid] = best; ri[tid] = bi;
        __syncthreads();
        for (int s = 128; s > 0; s >>= 1) {
            if (tid < s) {
                float ov = rv[tid + s]; int oi = ri[tid + s];
                if (ov < rv[tid] || (ov == rv[tid] && oi < ri[tid])) {
                    rv[tid] = ov; ri[tid] = oi;
                }
            }
            __syncthreads();
        }
        if (tid == 0) { sel[b * P_ + j] = ri[0]; se[ri[0]] = FLT_MAX; }
        __syncthreads();
    }
}

// ---------------- Kernel 4: WMMA attn rows for the selected 64, exp(d-lse) --
__global__ void attn_rows_kernel(const float* __restrict__ feat3,
                                 const int* __restrict__ sel,
                                 const float* __restrict__ lseg,
                                 float* __restrict__ attn_out) {
    __shared__ float As[C_ * 16];
    __shared__ int   selL[16];
    __shared__ float lseL[16];
    const int b = blockIdx.x / (P_ / 16);
    const int rt = blockIdx.x % (P_ / 16);
    const int rowbase = rt * 16;
    const float* X = feat3 + (size_t)b * C_ * N3_;
    if (threadIdx.x < 16) {
        int s = sel[b * P_ + rowbase + threadIdx.x];
        selL[threadIdx.x] = s;
        lseL[threadIdx.x] = lseg[(size_t)b * N3_ + s];
    }
    __syncthreads();
    for (int idx = threadIdx.x; idx < C_ * 16; idx += blockDim.x) {
        int c = idx >> 4, m = idx & 15;
        As[idx] = X[(size_t)c * N3_ + selL[m]];
    }
    __syncthreads();
    const int wave = threadIdx.x >> 5;
    const int lane = threadIdx.x & 31;
    const int g  = lane >> 4;
    const int lh = lane & 15;
    for (int st = wave; st < N3_ / 64; st += 4) {
        const int colbase = st * 64;
        {
            int nst = st + 4;
            if (nst < N3_ / 64) {
                const float* pf = X + nst * 64 + lane * 2;
                #pragma unroll
                for (int kk = 0; kk < C_; kk += 16)
                    __builtin_prefetch(pf + (size_t)kk * N3_, 0, 1);
            }
        }
        const float* Xc = X + colbase + lh;
        v8f acc0 = {}, acc1 = {}, acc2 = {}, acc3 = {};
        for (int k = 0; k < C_; k += 4) {
            v2f a;
            a.x = As[(k + 2 * g) * 16 + lh];
            a.y = As[(k + 2 * g + 1) * 16 + lh];
            const float* r0 = Xc + (size_t)(k + 2 * g) * N3_;
            const float* r1 = Xc + (size_t)(k + 2 * g + 1) * N3_;
            v2f b0, b1, b2, b3;
            b0.x = r0[0];  b0.y = r1[0];
            b1.x = r0[16]; b1.y = r1[16];
            b2.x = r0[32]; b2.y = r1[32];
            b3.x = r0[48]; b3.y = r1[48];
            acc0 = __builtin_amdgcn_wmma_f32_16x16x4_f32(false, a, false, b0, (short)0, acc0, false, false);
            acc1 = __builtin_amdgcn_wmma_f32_16x16x4_f32(false, a, false, b1, (short)0, acc1, false, false);
            acc2 = __builtin_amdgcn_wmma_f32_16x16x4_f32(false, a, false, b2, (short)0, acc2, false, false);
            acc3 = __builtin_amdgcn_wmma_f32_16x16x4_f32(false, a, false, b3, (short)0, acc3, false, false);
        }
        for (int r = 0; r < 8; ++r) {
            int row16 = r + 8 * g;
            float l = lseL[row16];
            size_t base = ((size_t)(b * P_) + rowbase + row16) * N3_ + colbase + lh;
            attn_out[base +  0] = __expf(acc0[r] - l);
            attn_out[base + 16] = __expf(acc1[r] - l);
            attn_out[base + 32] = __expf(acc2[r] - l);
            attn_out[base + 48] = __expf(acc3[r] - l);
        }
    }
}

// ---------------- Kernel 5: agg = attn_qs @ fr, then L2 normalize -----------
__global__ void agg_norm_kernel(const float* __restrict__ attn,
                                const float* __restrict__ feat3,
                                float* __restrict__ out) {
    __shared__ float arow[N3_];
    __shared__ float red[C_];
    int bp = blockIdx.x;
    int b = bp / P_;
    int c = threadIdx.x;
    const float* ar = attn + (size_t)bp * N3_;
    for (int i = c; i < N3_; i += C_) arow[i] = ar[i];
    __syncthreads();
    const float* xr = feat3 + (size_t)(b * C_ + c) * N3_;   // X[c][:] contiguous
    float sum = 0.f;
    for (int n = 0; n < N3_; n += 4) {
        float4 x = *(const float4*)(xr + n);
        sum += arow[n] * x.x + arow[n + 1] * x.y + arow[n + 2] * x.z + arow[n + 3] * x.w;
    }
    red[c] = sum * sum;
    __syncthreads();
    for (int s = C_ / 2; s > 0; s >>= 1) {
        if (c < s) red[c] += red[c + s];
        __syncthreads();
    }
    float norm = sqrtf(red[0]) + 1e-7f;
    out[(size_t)bp * C_ + c] = sum / norm;
}

extern "C" void kernel_launch(void* const* d_in, const int* in_sizes, int n_in,
                              void* d_out, int out_size, void* d_ws, size_t ws_size,
                              hipStream_t stream) {
    (void)in_sizes; (void)n_in; (void)out_size; (void)ws_size;
    const float* feat0 = (const float*)d_in[0];
    const float* feat1 = (const float*)d_in[1];
    const float* feat2 = (const float*)d_in[2];
    const float* feat3 = (const float*)d_in[3];
    const int*   pid0  = (const int*)d_in[4];
    const int*   pid1  = (const int*)d_in[5];
    const int*   pid2  = (const int*)d_in[6];
    float* out = (float*)d_out;

    float* entg = (float*)d_ws;                 // [2][4096]
    float* lseg = entg + B_ * N3_;              // [2][4096]
    int*   sel  = (int*)(lseg + B_ * N3_);      // [2][64]

    const size_t xrow = (size_t)B_ * P_ * C_;   // 128*256 per stack slice
    float* xs0  = out;
    float* xs1  = out + xrow;
    float* xs2  = out + 2 * xrow;
    float* xs3  = out + 3 * xrow;
    float* attn = out + 4 * xrow;               // attn_qs [2][64][4096]

    gather_norm_kernel<<<B_ * P_, C_, 0, stream>>>(feat0, pid0, xs0, 32 * 32 * 16);
    gather_norm_kernel<<<B_ * P_, C_, 0, stream>>>(feat1, pid1, xs1, 24 * 24 * 16);
    gather_norm_kernel<<<B_ * P_, C_, 0, stream>>>(feat2, pid2, xs2, 16 * 16 * 16);

    entropy_kernel<<<B_ * (N3_ / 16), 128, 0, stream>>>(feat3, entg, lseg);
    select_kernel<<<B_, 256, 0, stream>>>(entg, sel);
    attn_rows_kernel<<<B_ * (P_ / 16), 128, 0, stream>>>(feat3, sel, lseg, attn);
    agg_norm_kernel<<<B_ * P_, C_, 0, stream>>>(attn, feat3, xs3);
}